// Cos_loss_60370060312864
// MI455X (gfx1250) — compile-verified
//
#include <hip/hip_runtime.h>
#include <hip/hip_bf16.h>

typedef unsigned int u32;
typedef unsigned long long u64;
typedef u32 v4u __attribute__((ext_vector_type(4)));
typedef int  v8i __attribute__((ext_vector_type(8)));
typedef int  v4i __attribute__((ext_vector_type(4)));

// Geometry fixed by the reference: n=256, t=512, 15 joints * 9 f32 per row.
constexpr u32 ROW_FLOATS   = 135;              // 15*9
constexpr u32 ROW_BYTES    = 540;
constexpr u32 SEGB_BYTE    = 396;              // joint 11 starts at float 99
constexpr u32 N_ROWS       = 256u * 512u;      // 131072 (n*t)
constexpr u32 NBLOCKS      = 512;
constexpr u32 ROWS_PER_BLK = N_ROWS / NBLOCKS; // 256
constexpr u32 TILE_ROWS    = 64;
constexpr u32 NTILES       = ROWS_PER_BLK / TILE_ROWS; // 4
constexpr u32 SEG_FLOATS   = TILE_ROWS * 18;   // 1152 (18 f32 per row: joint pair)
constexpr u32 REGION       = SEG_FLOATS + 3;   // +3 f32 pad -> disjoint LDS bank sets
constexpr u32 BUF_FLOATS   = 4 * REGION;       // xA, xB, yA, yB
constexpr float INV_COUNT  = 1.0f / (float)(N_ROWS * 4u);

// One TDM 2D-tile load: 18 f32 per row, TILE_ROWS rows, row stride 135 f32,
// packed contiguously into LDS at lds_byte_addr. Descriptor per ISA ch.8.
__device__ __forceinline__ void tdm_load_tile18(u64 gaddr, u32 lds_byte_addr) {
  v4u g0;
  g0[0] = 1u;                                    // count=1 valid user descriptor
  g0[1] = lds_byte_addr;                         // D#.lds_addr (bytes)
  g0[2] = (u32)(gaddr & 0xFFFFFFFFull);          // D#.global_addr[31:0]
  g0[3] = (u32)((gaddr >> 32) & 0x01FFFFFFull)   // D#.global_addr[56:32]
        | (2u << 30);                            // D#.type = 2 ("image")
  const u32 TDIM1 = 0x00100000u;                 // huge dim1: never clamps rows
  v8i g1;
  g1[0] = (int)(2u << 16);                       // data_size=2 (4B); no multicast/pad
  g1[1] = (int)((ROW_FLOATS & 0xFFFFu) << 16);   // tensor_dim0[15:0] = 135
  g1[2] = (int)(((ROW_FLOATS >> 16) & 0xFFFFu)   // tensor_dim0[31:16]
        |       ((TDIM1 & 0xFFFFu) << 16));      // tensor_dim1[15:0]
  g1[3] = (int)(((TDIM1 >> 16) & 0xFFFFu)        // tensor_dim1[31:16]
        |       (18u << 16));                    // tile_dim0 = 18 elements
  g1[4] = (int)(TILE_ROWS & 0xFFFFu);            // tile_dim1 = 64; tile_dim2 = 0
  g1[5] = (int)ROW_FLOATS;                       // tensor_dim0_stride[31:0] = 135
  g1[6] = 0;                                     // stride0 hi / stride1 lo
  g1[7] = 0;                                     // stride1 hi
  v4i g2 = {0, 0, 0, 0};                         // pure 2D tile: groups 2/3 unused
  v4i g3 = {0, 0, 0, 0};
  v8i g4 = {0, 0, 0, 0, 0, 0, 0, 0};             // extra group (6-arg toolchain form)
  __builtin_amdgcn_tensor_load_to_lds(g0, g1, g2, g3, g4, 0);
}

__device__ __forceinline__ void issue_tile(const float* x, const float* y,
                                           u32 row0, u32 lds_buf_byte) {
  u64 xb = (u64)x + (u64)row0 * ROW_BYTES;
  u64 yb = (u64)y + (u64)row0 * ROW_BYTES;
  tdm_load_tile18(xb,             lds_buf_byte + 0u * REGION * 4u); // x joints 0,1
  tdm_load_tile18(xb + SEGB_BYTE, lds_buf_byte + 1u * REGION * 4u); // x joints 11,12
  tdm_load_tile18(yb,             lds_buf_byte + 2u * REGION * 4u); // y joints 0,1
  tdm_load_tile18(yb + SEGB_BYTE, lds_buf_byte + 3u * REGION * 4u); // y joints 11,12
}

__global__ void __launch_bounds__(256)
cos_loss_partial(const float* __restrict__ x, const float* __restrict__ y,
                 float* __restrict__ partial) {
  __shared__ float smem[2 * BUF_FLOATS];   // double-buffered tile stage
  __shared__ float wred[8];
  const u32 tid  = threadIdx.x;
  const u32 row0 = blockIdx.x * ROWS_PER_BLK;
  const u32 lds0 = (u32)(u64)&smem[0];     // flat->LDS: low 32 bits are LDS offset
  const bool wave0 = (tid < 32u);          // wave-uniform; TDM issued once per WG

  if (wave0) issue_tile(x, y, row0, lds0); // prefetch tile 0 into buffer 0

  float acc = 0.0f;
  for (u32 t = 0; t < NTILES; ++t) {
    if (wave0) {
      if (t + 1 < NTILES) {
        issue_tile(x, y, row0 + (t + 1) * TILE_ROWS,
                   lds0 + ((t + 1) & 1u) * BUF_FLOATS * 4u);
        __builtin_amdgcn_s_wait_tensorcnt(4); // tile t done; tile t+1 in flight
      } else {
        __builtin_amdgcn_s_wait_tensorcnt(0); // drain
      }
    }
    __syncthreads();                          // TDM data visible to all waves

    const float* buf = &smem[(t & 1u) * BUF_FLOATS];
    const u32 row = tid >> 2;                 // 64 rows * 4 joints = 256 threads
    const u32 j   = tid & 3u;                 // 0,1 -> seg A; 2,3 -> seg B
    const u32 off = row * 18u + (j & 1u) * 9u;
    const float* xp = buf + (j >> 1) * REGION + off;
    const float* yp = buf + 2u * REGION + (j >> 1) * REGION + off;
    float tr = 0.0f;
#pragma unroll
    for (int k = 0; k < 9; ++k) tr = __builtin_fmaf(xp[k], yp[k], tr);
    float tmp = fminf(tr - 1.0f, 2.0f);       // reference clamps upper bound only
    acc += fabsf(acosf(0.5f * tmp));

    __syncthreads();                          // buffer reusable before next issue
  }

  // wave32 reduction, then 8 wave-partials -> block partial (deterministic)
#pragma unroll
  for (int o = 16; o > 0; o >>= 1) acc += __shfl_down(acc, o, 32);
  if ((tid & 31u) == 0u) wred[tid >> 5] = acc;
  __syncthreads();
  if (tid == 0) {
    float s = 0.0f;
#pragma unroll
    for (int w = 0; w < 8; ++w) s += wred[w];
    partial[blockIdx.x] = s;
  }
}

__global__ void __launch_bounds__(256)
cos_loss_finalize(const float* __restrict__ partial, float* __restrict__ out) {
  __shared__ float red[256];
  const u32 tid = threadIdx.x;
  red[tid] = partial[tid] + partial[tid + 256];
  __syncthreads();
#pragma unroll
  for (int o = 128; o > 0; o >>= 1) {
    if (tid < (u32)o) red[tid] += red[tid + o];
    __syncthreads();
  }
  if (tid == 0) out[0] = red[0] * INV_COUNT;
}

extern "C" void kernel_launch(void* const* d_in, const int* in_sizes, int n_in,
                              void* d_out, int out_size, void* d_ws, size_t ws_size,
                              hipStream_t stream) {
  (void)in_sizes; (void)n_in; (void)out_size; (void)ws_size;
  const float* x = (const float*)d_in[0];
  const float* y = (const float*)d_in[1];
  float* partial = (float*)d_ws;       // 512 floats of scratch, fully overwritten
  float* out     = (float*)d_out;
  cos_loss_partial<<<NBLOCKS, 256, 0, stream>>>(x, y, partial);
  cos_loss_finalize<<<1, 256, 0, stream>>>(partial, out);
}